// MT_ResAMIL_BagIns_35184372089510
// MI455X (gfx1250) — compile-verified
//
#include <hip/hip_runtime.h>
#include <hip/hip_bf16.h>
#include <math.h>

typedef __attribute__((ext_vector_type(2))) float v2f;
typedef __attribute__((ext_vector_type(8))) float v8f;

#define N_INST 131072
#define LFEAT  512
#define BAGS   512
#define NCLS   128

// ---------------------------------------------------------------------------
// Kernel 1: per-bag attention softmax + weighted row-sum of H.
// One block (256 threads) per bag; batch[] is sorted so each bag is a
// contiguous range found by binary search. Deterministic block reductions.
// H is streamed exactly once (256 MB) -> HBM-bound.
// ---------------------------------------------------------------------------
__global__ __launch_bounds__(256)
void bag_pool_kernel(const float* __restrict__ H,
                     const float* __restrict__ att,
                     const int*   __restrict__ batch,
                     float*       __restrict__ bagFeat) {
  const int b = blockIdx.x;
  const int t = threadIdx.x;           // 0..255

  __shared__ int   sRange[2];
  __shared__ float sRed[256];
  __shared__ float sW[256];

  if (t == 0) {
    // lower_bound(b)
    int lo = 0, hi = N_INST;
    while (lo < hi) { int mid = (lo + hi) >> 1; if (batch[mid] <  b) lo = mid + 1; else hi = mid; }
    sRange[0] = lo;
    // lower_bound(b+1)
    hi = N_INST;
    while (lo < hi) { int mid = (lo + hi) >> 1; if (batch[mid] <= b) lo = mid + 1; else hi = mid; }
    sRange[1] = lo;
  }
  __syncthreads();
  const int start = sRange[0];
  const int end   = sRange[1];

  // ---- per-bag max of attention -------------------------------------------
  float m = -3.402823466e38f;
  for (int i = start + t; i < end; i += 256) m = fmaxf(m, att[i]);
  sRed[t] = m;
  __syncthreads();
  for (int s = 128; s > 0; s >>= 1) {
    if (t < s) sRed[t] = fmaxf(sRed[t], sRed[t + s]);
    __syncthreads();
  }
  const float bagMax = sRed[0];
  __syncthreads();

  // ---- per-bag sum of exp --------------------------------------------------
  float sum = 0.0f;
  for (int i = start + t; i < end; i += 256) sum += expf(att[i] - bagMax);
  sRed[t] = sum;
  __syncthreads();
  for (int s = 128; s > 0; s >>= 1) {
    if (t < s) sRed[t] += sRed[t + s];
    __syncthreads();
  }
  const float denom = sRed[0];
  const float inv   = (denom > 0.0f) ? (1.0f / denom) : 0.0f;
  __syncthreads();

  // ---- weighted accumulation of member rows -------------------------------
  // Thread t owns feature columns 2t and 2t+1 (float2, coalesced b64 loads).
  float2 acc = make_float2(0.0f, 0.0f);
  for (int cs = start; cs < end; cs += 256) {
    const int cnt = min(256, end - cs);
    if (t < cnt) sW[t] = expf(att[cs + t] - bagMax) * inv;
    __syncthreads();
    #pragma unroll 4
    for (int j = 0; j < cnt; ++j) {
      const float  w   = sW[j];
      const float2 row = *(const float2*)(H + (size_t)(cs + j) * LFEAT + 2 * t);
      acc.x = fmaf(w, row.x, acc.x);
      acc.y = fmaf(w, row.y, acc.y);
    }
    __syncthreads();
  }
  *(float2*)(bagFeat + (size_t)b * LFEAT + 2 * t) = acc;
}

// ---------------------------------------------------------------------------
// Kernel 2: out(512x128) = bagFeat(512x512) @ fc1_w^T(512x128) + bias
// One wave per 16x16 output tile, V_WMMA_F32_16X16X4_F32, K-loop step 4.
// A-layout (32-bit 16x4): lanes 0-15 hold K=0,1 in v[0],v[1]; lanes 16-31 K=2,3.
// C/D layout: VGPR r -> row m0 + 8*(lane>=16) + r, column = n0 + (lane&15).
// ---------------------------------------------------------------------------
__global__ __launch_bounds__(256)
void out_gemm_wmma(const float* __restrict__ bagFeat,
                   const float* __restrict__ W,      // (NCLS, LFEAT) row-major
                   const float* __restrict__ bias,   // (NCLS)
                   float*       __restrict__ out) {  // (BAGS, NCLS)
  const int lane  = threadIdx.x & 31;
  const int wave  = threadIdx.x >> 5;
  const int gwave = blockIdx.x * (blockDim.x >> 5) + wave;   // 0..255

  const int mTile = gwave >> 3;        // 32 tiles along M (bags)
  const int nTile = gwave & 7;         // 8 tiles along N (classes)
  const int m0 = mTile * 16;
  const int n0 = nTile * 16;

  const int half = lane >> 4;          // 0: K=0,1   1: K=2,3
  const int l16  = lane & 15;

  // accumulator seeded with bias (broadcast down each output column)
  const float bv = bias[n0 + l16];
  v8f c;
  #pragma unroll
  for (int r = 0; r < 8; ++r) c[r] = bv;

  const float* aRow = bagFeat + (size_t)(m0 + l16) * LFEAT;  // A row per lane
  const float* bRow = W       + (size_t)(n0 + l16) * LFEAT;  // B^T col = W row

  #pragma unroll 8
  for (int k = 0; k < LFEAT; k += 4) {
    const int ka = k + 2 * half;
    const v2f a = *(const v2f*)(aRow + ka);
    const v2f b = *(const v2f*)(bRow + ka);
    c = __builtin_amdgcn_wmma_f32_16x16x4_f32(
            /*neg_a=*/false, a, /*neg_b=*/false, b,
            /*c_mod=*/(short)0, c, /*reuse_a=*/false, /*reuse_b=*/false);
  }

  #pragma unroll
  for (int r = 0; r < 8; ++r) {
    const int mm = m0 + half * 8 + r;
    out[(size_t)mm * NCLS + n0 + l16] = c[r];
  }
}

extern "C" void kernel_launch(void* const* d_in, const int* in_sizes, int n_in,
                              void* d_out, int out_size, void* d_ws, size_t ws_size,
                              hipStream_t stream) {
  const float* H     = (const float*)d_in[0];  // (N, L)
  const float* att   = (const float*)d_in[1];  // (1, N)
  const int*   batch = (const int*)  d_in[2];  // (N,) sorted bag ids
  const float* fc1_w = (const float*)d_in[3];  // (NUM_CLASSES, L)
  const float* fc1_b = (const float*)d_in[4];  // (NUM_CLASSES,)
  float* out = (float*)d_out;                  // (BAG_NUM, NUM_CLASSES)

  float* bagFeat = (float*)d_ws;               // 512*512 f32 = 1 MB scratch

  bag_pool_kernel<<<BAGS, 256, 0, stream>>>(H, att, batch, bagFeat);
  out_gemm_wmma<<<32, 256, 0, stream>>>(bagFeat, fc1_w, fc1_b, out);
}